// Subtest_31318901522626
// MI455X (gfx1250) — compile-verified
//
#include <hip/hip_runtime.h>
#include <math.h>

typedef __attribute__((ext_vector_type(2))) float v2f;
typedef __attribute__((ext_vector_type(8))) float v8f;

namespace {

constexpr int kL   = 64;    // digits per sequence
constexpr int kK   = 16;    // base
constexpr int kTPB = 256;   // 8 wave32 waves
constexpr int kBatchPerWave  = 16;
constexpr int kBatchPerBlock = (kTPB / 32) * kBatchPerWave;   // 128

__device__ __forceinline__ v2f mk2(float x, float y) { v2f r; r[0] = x; r[1] = y; return r; }

// D = A(16x4) * B(4x16) + C, fp32 WMMA (full precision).
__device__ __forceinline__ v8f wmma4(v2f a, v2f b, v8f c) {
  return __builtin_amdgcn_wmma_f32_16x16x4_f32(false, a, false, b, (short)0, c, false, false);
}

// Swap values between lane L and lane L^16 (SWAPX16 swizzle: xor=0x10, and=0x1f).
__device__ __forceinline__ float swapx16(float v) {
  return __int_as_float(__builtin_amdgcn_ds_swizzle(__float_as_int(v), 0x401f));
}

// 32B global -> LDS for this lane via CDNA5 async loads (ASYNCcnt-tracked).
__device__ __forceinline__ void async32(unsigned lds_off, unsigned long long g) {
  asm volatile("global_load_async_to_lds_b128 %0, %1, off"           :: "v"(lds_off), "v"(g) : "memory");
  asm volatile("global_load_async_to_lds_b128 %0, %1, off offset:16" :: "v"(lds_off), "v"(g) : "memory");
}

__global__ __launch_bounds__(kTPB) void dft16_sub_scan(
    const float* __restrict__ op1,
    const float* __restrict__ op2,
    float* __restrict__ out)
{
  // [double-buf][local batch][ a(16) | c(16) ]  = 32 KB
  __shared__ float sbuf[2][kBatchPerBlock][2 * kK];

  const int tid  = threadIdx.x;
  const int wave = tid >> 5;
  const int lane = tid & 31;
  const int m    = lane & 15;   // batch slot (lower/upper half pair share a batch)
  const int h    = lane >> 4;   // lane-half: also selects contraction sub-rows

  const int lb = wave * kBatchPerWave + m;                 // local batch id
  const long long gb = (long long)blockIdx.x * kBatchPerBlock + lb;

  const unsigned long long gA = (unsigned long long)(op1 + gb * (kL * kK)) + (unsigned)(h * 32);
  const unsigned long long gC = (unsigned long long)(op2 + gb * (kL * kK)) + (unsigned)(h * 32);
  float* __restrict__ gw = out + ((long long)blockIdx.x * kBatchPerBlock + wave * kBatchPerWave) * (kL * kK);

  const unsigned rowOff    = (unsigned)(size_t)(&sbuf[0][lb][0]);
  const unsigned bufStride = (unsigned)(kBatchPerBlock * 2 * kK * sizeof(float));
  const unsigned dstA = rowOff + (unsigned)(h * 32);
  const unsigned dstC = rowOff + 64u + (unsigned)(h * 32);

  // Prime the async pipeline with step 0 (each lane stages half of its batch row).
  async32(dstA, gA);
  async32(dstC, gC);

  // ---------------- shared-matrix constants (per-lane register images) ----------
  // WMMA f32 16x16x4 operand layouts: A[m',k'] : lane=m', k' = 2h + elem.
  //                                   B[k',n ] : lane=n,  k' = 2h + elem.
  //                                   D[m ,n ] : VGPR j -> m=j+8h, lane -> n.
  const float th = 6.283185307179586f / 16.f;
  v2f mc[4], ms[4], uT[4], iT[4], wc3[4], ws3[4];
  float ckk[8], skk[8];
  const float hs = h ? -1.f : 1.f;
#pragma unroll
  for (int c2 = 0; c2 < 4; ++c2) {
    const int x0 = 4 * c2 + 2 * h;         // contraction index for elem 0
    // Stage-1 A operands: DFT row matrices (M = bin k, K = x):
    mc[c2] = mk2(cosf(th * m * x0), cosf(th * m * (x0 + 1)));       //  cos(th*k*x)
    ms[c2] = mk2(-sinf(th * m * x0), -sinf(th * m * (x0 + 1)));     // -sin(th*k*x)
    // Strict upper-triangular (exclusive prefix) and identity (transpose):
    uT[c2] = mk2(x0 < m ? 1.f : 0.f, (x0 + 1) < m ? 1.f : 0.f);
    iT[c2] = mk2(x0 == m ? 1.f : 0.f, (x0 + 1) == m ? 1.f : 0.f);
    // Stage-3 B operands: IDFT matrices with rows permuted to sigma_c = (2c,2c+1,2c+8,2c+9)
    // because stage-2 D-registers reinterpret as A with that row order.
    const int r0 = 2 * c2 + 8 * h;
    wc3[c2] = mk2(cosf(th * r0 * m) * (1.f / 16.f), cosf(th * (r0 + 1) * m) * (1.f / 16.f));
    ws3[c2] = mk2(-sinf(th * r0 * m) * (1.f / 16.f), -sinf(th * (r0 + 1) * m) * (1.f / 16.f));
  }
#pragma unroll
  for (int j = 0; j < 8; ++j) {            // k = j + 8h ; cos(th*(j+8)) = -cos(th*j)
    ckk[j] = hs * cosf(th * j);
    skk[j] = hs * sinf(th * j);
  }

  float b0 = 1.f, b1 = 0.f;                // borrow distribution, replicated in both lane halves

  for (int l = 0; l < kL; ++l) {
    const int buf = l & 1;
    if (l + 1 < kL) {
      const unsigned nb = (unsigned)(buf ^ 1) * bufStride;
      async32(dstA + nb, gA + (unsigned long long)((l + 1) * 64));
      async32(dstC + nb, gC + (unsigned long long)((l + 1) * 64));
      // async loads complete in order: cnt<=4 => this step's 4 staged lines landed.
      asm volatile("s_wait_asynccnt 4" ::: "memory");
    } else {
      asm volatile("s_wait_asynccnt 0" ::: "memory");
    }

    // B-operand data chunks straight from LDS: elem pair = floats (4c+2h, 4c+2h+1).
    const float* row = &sbuf[buf][lb][0];
    v2f aB[4], cB[4];
#pragma unroll
    for (int c2 = 0; c2 < 4; ++c2) {
      aB[c2] = *(const v2f*)(row + 4 * c2 + 2 * h);
      cB[c2] = *(const v2f*)(row + kK + 4 * c2 + 2 * h);
    }

    // ---- Stage 1: shared-matrix GEMMs (all contract over x, K=16 in 4 chunks) ----
    // FaT[k,b], FcT[k,b] = DFT of a,c ; ApreT[y,b] = exclusive prefix ; ceT[y,b] = c[b,y].
    v8f faR = {}, faI = {}, fcR = {}, fcI = {}, apre = {}, ce = {};
#pragma unroll
    for (int c2 = 0; c2 < 4; ++c2) {
      faR  = wmma4(mc[c2], aB[c2], faR);
      faI  = wmma4(ms[c2], aB[c2], faI);
      fcR  = wmma4(mc[c2], cB[c2], fcR);
      fcI  = wmma4(ms[c2], cB[c2], fcI);
      apre = wmma4(uT[c2], aB[c2], apre);
      ce   = wmma4(iT[c2], cB[c2], ce);
    }

    // ---- Stage 2: P = Fa * conj(Fc), elementwise in the (k,b) D-layout ----
    v8f PR, PI;
    float sPR = 0.f, sDot = 0.f;
#pragma unroll
    for (int j = 0; j < 8; ++j) {
      PR[j] = faR[j] * fcR[j] + faI[j] * fcI[j];
      PI[j] = faI[j] * fcR[j] - faR[j] * fcI[j];
      sPR  += PR[j];                 // Parseval: sum_k PR = 16 * dot(a,c)
      sDot += apre[j] * ce[j];       // sum_y Apre[y]*c[y] (this half's k-range)
    }
    const float dotAC = (sPR + swapx16(sPR)) * (1.f / 16.f);
    const float s_lt  = sDot + swapx16(sDot);          // sum_{x<y} a_x c_y
    const float s_le  = s_lt + dotAC;                  // sum_{x<=y} a_x c_y
    float suma = faR[0]; { const float t = swapx16(suma); suma = h ? t : suma; }  // Fa[k=0]
    float sumc = fcR[0]; { const float t = swapx16(sumc); sumc = h ? t : sumc; }  // Fc[k=0]

    // ---- Borrow mix in frequency domain: Q[k] = P[k] * (b0 + b1*e^{i*th*k}) ----
    v8f QR, QI;
#pragma unroll
    for (int j = 0; j < 8; ++j) {
      const float t0 = b0 + b1 * ckk[j];
      const float t1 = b1 * skk[j];
      QR[j] = PR[j] * t0 - PI[j] * t1;
      QI[j] = PI[j] * t0 + PR[j] * t1;
    }

    // ---- Stage 3: res = IDFT(Q). D-registers of Q reinterpret directly as A-operands
    //      (row order sigma_c), B operands are the pre-permuted IDFT constants. ----
    v8f resD = {};
#pragma unroll
    for (int c2 = 0; c2 < 4; ++c2) {
      resD = wmma4(mk2(QR[2 * c2], QR[2 * c2 + 1]), wc3[c2], resD);
      resD = wmma4(mk2(QI[2 * c2], QI[2 * c2 + 1]), ws3[c2], resD);
    }

    // resD: VGPR j -> batch (j + 8h), lane -> digit d = m. 16 lanes * b32 = 64B rows.
#pragma unroll
    for (int j = 0; j < 8; ++j)
      gw[(long long)(j + 8 * h) * (kL * kK) + l * kK + m] = resD[j];

    // ---- Borrow recurrence (replicated across lane halves) ----
    const float p1 = b0 * s_lt + b1 * s_le;
    const float tt = suma * sumc * (b0 + b1);
    b0 = tt - p1;
    b1 = p1;
  }
}

} // namespace

extern "C" void kernel_launch(void* const* d_in, const int* in_sizes, int n_in,
                              void* d_out, int out_size, void* d_ws, size_t ws_size,
                              hipStream_t stream) {
  const float* op1 = (const float*)d_in[0];
  const float* op2 = (const float*)d_in[1];
  float* out = (float*)d_out;
  const int n_batch = in_sizes[0] / (kL * kK);              // 65536
  const int blocks  = n_batch / kBatchPerBlock;             // 512 (exact for ref shapes)
  dft16_sub_scan<<<blocks, kTPB, 0, stream>>>(op1, op2, out);
}